// LGCNEncoder_1589137900101
// MI455X (gfx1250) — compile-verified
//
#include <hip/hip_runtime.h>
#include <hip/hip_bf16.h>
#include <stdint.h>
#include <stddef.h>

// ---------------- problem constants (match reference) ----------------
#define USERS    100000
#define ITEMS    50000
#define NODES    150000       // USERS + ITEMS
#define EMB      64
#define NNZ_CNT  4800000
#define QBATCH   4096

// ---------------- CDNA5 async global->LDS path ----------------
#if defined(__HIP_DEVICE_COMPILE__) && __has_builtin(__builtin_amdgcn_global_load_async_to_lds_b128)
#define HAVE_ASYNC_LDS 1
#endif

#if defined(__HIP_DEVICE_COMPILE__) && __has_builtin(__builtin_amdgcn_s_wait_asynccnt)
#define WAIT_ASYNC(n) __builtin_amdgcn_s_wait_asynccnt(n)
#else
#define WAIT_ASYNC(n) asm volatile("s_wait_asynccnt " #n ::: "memory")
#endif

#ifdef HAVE_ASYNC_LDS
typedef int v4i_t __attribute__((vector_size(16)));
// param 0: v4i __device__* (AS1 global source)  -- from hipcc diagnostic
// param 1: v4i __shared__* (AS3 LDS destination)
#define ASYNC_LDS_B128(gp, lp)                                        \
  __builtin_amdgcn_global_load_async_to_lds_b128(                     \
      (__attribute__((address_space(1))) v4i_t*)(gp),                 \
      (__attribute__((address_space(3))) v4i_t*)(lp), 0, 0)
#endif

// ---------------- kernels ----------------

__global__ void lgcn_concat(const float* __restrict__ ue,
                            const float* __restrict__ ie,
                            float* __restrict__ xa) {
  int i = blockIdx.x * blockDim.x + threadIdx.x;       // float4 index
  const int NU4 = USERS * EMB / 4;
  const int NT4 = NODES * EMB / 4;
  if (i >= NT4) return;
  float4 v;
  if (i < NU4) v = ((const float4*)ue)[i];
  else         v = ((const float4*)ie)[i - NU4];
  ((float4*)xa)[i] = v;
}

__global__ void lgcn_zero_i32(int* __restrict__ p, int n) {
  int i = blockIdx.x * blockDim.x + threadIdx.x;
  if (i < n) p[i] = 0;
}

__global__ void lgcn_hist(const int* __restrict__ row, int* __restrict__ cnt, int nnz) {
  int e = blockIdx.x * blockDim.x + threadIdx.x;
  if (e < nnz) atomicAdd(&cnt[row[e]], 1);
}

__global__ __launch_bounds__(1024) void lgcn_scan(int* __restrict__ counts,
                                                  int* __restrict__ row_ptr,
                                                  int n) {
  __shared__ int sm[1024];
  __shared__ int carry;
  int tid = threadIdx.x;
  if (tid == 0) carry = 0;
  __syncthreads();
  for (int base = 0; base < n; base += 1024) {
    int i = base + tid;
    int v = (i < n) ? counts[i] : 0;
    sm[tid] = v;
    __syncthreads();
    for (int off = 1; off < 1024; off <<= 1) {
      int t = (tid >= off) ? sm[tid - off] : 0;
      __syncthreads();
      sm[tid] += t;
      __syncthreads();
    }
    int incl = sm[tid];
    int c = carry;
    if (i < n) {
      row_ptr[i + 1] = c + incl;
      counts[i]      = c + incl - v;   // exclusive = row start (becomes cursor)
    }
    __syncthreads();
    if (tid == 1023) carry = c + sm[1023];
    __syncthreads();
  }
  if (tid == 0) row_ptr[0] = 0;
}

__global__ void lgcn_scatter(const int* __restrict__ row,
                             const int* __restrict__ col,
                             const float* __restrict__ val,
                             int* __restrict__ cursor,
                             int* __restrict__ csr_col,
                             float* __restrict__ csr_val, int nnz) {
  int e = blockIdx.x * blockDim.x + threadIdx.x;
  if (e >= nnz) return;
  int r = row[e];
  int p = atomicAdd(&cursor[r], 1);
  csr_col[p] = col[e];
  csr_val[p] = val[e];
}

__global__ __launch_bounds__(256) void lgcn_spmm(const float* __restrict__ xin,
                                                 float* __restrict__ xout,
                                                 const int* __restrict__ row_ptr,
                                                 const int* __restrict__ csr_col,
                                                 const float* __restrict__ csr_val) {
  const int wave = threadIdx.x >> 5;
  const int lane = threadIdx.x & 31;
  const int rowi = blockIdx.x * 8 + wave;
  if (rowi >= NODES) return;

  const int s = row_ptr[rowi];
  const int e = row_ptr[rowi + 1];
  float2 acc; acc.x = 0.f; acc.y = 0.f;
  int j = s;

#ifdef HAVE_ASYNC_LDS
  __shared__ float smem[8][2][16 * EMB];   // 8 waves x 2 buffers x 4KB = 64KB
  const int CH = 16;
  const int nfull = (e - s) / CH;
  const int l15  = lane & 15;
  const int half = lane >> 4;

  auto stage = [&](int c, int b) {
    int base = s + c * CH;
    int colv = csr_col[base + l15];             // lanes 0..15 hold the 16 cols
    #pragma unroll
    for (int t = 0; t < 8; ++t) {
      int n  = 2 * t + half;                    // neighbor slot 0..15
      int cn = __shfl(colv, n, 32);
      const float* gp = xin + (size_t)cn * EMB + (l15 << 2);   // 16B per lane
      float*       lp = &smem[wave][b][n * EMB + (l15 << 2)];
      ASYNC_LDS_B128(gp, lp);                   // 32 lanes x 16B = rows 2t,2t+1
    }
  };
  auto consume = [&](int c, int b) {
    int base = s + c * CH;
    float valv = csr_val[base + l15];
    #pragma unroll
    for (int n = 0; n < CH; ++n) {
      float v = __shfl(valv, n, 32);
      float2 d = *(const float2*)&smem[wave][b][n * EMB + lane * 2];
      acc.x += v * d.x;
      acc.y += v * d.y;
    }
  };

  if (nfull > 0) {
    stage(0, 0);
    for (int c = 0; c < nfull; ++c) {
      if (c + 1 < nfull) { stage(c + 1, (c + 1) & 1); WAIT_ASYNC(8); }
      else               { WAIT_ASYNC(0); }
      consume(c, c & 1);
    }
    j = s + nfull * CH;
  }
#endif

  for (; j < e; ++j) {
    if (j + 8 < e) __builtin_prefetch(xin + (size_t)csr_col[j + 8] * EMB, 0, 0);
    int   cn = csr_col[j];
    float v  = csr_val[j];
    float2 d = *(const float2*)(xin + (size_t)cn * EMB + lane * 2);
    acc.x += v * d.x;
    acc.y += v * d.y;
  }

  *(float2*)(xout + (size_t)rowi * EMB + lane * 2) = acc;
}

__global__ void lgcn_gather(const float* __restrict__ x,
                            const int* __restrict__ users,
                            const int* __restrict__ items,
                            float* __restrict__ out,
                            float alpha, int accumulate) {
  int t = blockIdx.x * blockDim.x + threadIdx.x;
  if (t >= 2 * QBATCH * EMB) return;
  int r = t >> 6;
  int d = t & 63;
  int node = (r < QBATCH) ? users[r] : (USERS + items[r - QBATCH]);
  float v = alpha * x[(size_t)node * EMB + d];
  out[t] = (accumulate ? out[t] : 0.f) + v;
}

// ---------------- launcher ----------------
extern "C" void kernel_launch(void* const* d_in, const int* in_sizes, int n_in,
                              void* d_out, int out_size, void* d_ws, size_t ws_size,
                              hipStream_t stream) {
  (void)in_sizes; (void)n_in; (void)out_size; (void)ws_size;
  const float* user_emb = (const float*)d_in[0];
  const float* item_emb = (const float*)d_in[1];
  const int*   adj_row  = (const int*)d_in[2];
  const int*   adj_col  = (const int*)d_in[3];
  const float* adj_val  = (const float*)d_in[4];
  const int*   users    = (const int*)d_in[5];
  const int*   items    = (const int*)d_in[6];
  float* out = (float*)d_out;

  char* ws = (char*)d_ws;
  float* xa      = (float*)(ws + 0);
  float* xb      = (float*)(ws + 38400000);
  int*   row_ptr = (int*)  (ws + 76800000);
  int*   cursor  = (int*)  (ws + 77400064);
  int*   csr_col = (int*)  (ws + 78000128);
  float* csr_val = (float*)(ws + 97200128);

  const int T = 256;
  lgcn_concat<<<(NODES * EMB / 4 + T - 1) / T, T, 0, stream>>>(user_emb, item_emb, xa);
  lgcn_zero_i32<<<(NODES + T - 1) / T, T, 0, stream>>>(cursor, NODES);
  lgcn_hist<<<(NNZ_CNT + T - 1) / T, T, 0, stream>>>(adj_row, cursor, NNZ_CNT);
  lgcn_scan<<<1, 1024, 0, stream>>>(cursor, row_ptr, NODES);
  lgcn_scatter<<<(NNZ_CNT + T - 1) / T, T, 0, stream>>>(adj_row, adj_col, adj_val,
                                                        cursor, csr_col, csr_val, NNZ_CNT);
  const int GT = (2 * QBATCH * EMB + T - 1) / T;
  lgcn_gather<<<GT, T, 0, stream>>>(xa, users, items, out, 0.25f, 0);
  const int SB = (NODES + 7) / 8;
  lgcn_spmm<<<SB, T, 0, stream>>>(xa, xb, row_ptr, csr_col, csr_val);
  lgcn_gather<<<GT, T, 0, stream>>>(xb, users, items, out, 0.25f, 1);
  lgcn_spmm<<<SB, T, 0, stream>>>(xb, xa, row_ptr, csr_col, csr_val);
  lgcn_gather<<<GT, T, 0, stream>>>(xa, users, items, out, 0.25f, 1);
  lgcn_spmm<<<SB, T, 0, stream>>>(xa, xb, row_ptr, csr_col, csr_val);
  lgcn_gather<<<GT, T, 0, stream>>>(xb, users, items, out, 0.25f, 1);
}